// MetricLoss_53231824666931
// MI455X (gfx1250) — compile-verified
//
#include <hip/hip_runtime.h>

typedef __attribute__((ext_vector_type(2))) float v2f;
typedef __attribute__((ext_vector_type(8))) float v8f;

#define BB   1024
#define MM   32
#define FF   256
#define KGRP 4

// ---------------------------------------------------------------------------
// Kernel 1: sq[b*MM+m] = sum_f x[b,m,f]^2   (one wave32 per row)
// Block 0 also zero-initializes the two output scalars (stream order
// guarantees this happens before kernel 2's atomics).
// ---------------------------------------------------------------------------
__global__ void MetricLoss_sq_kernel(const float* __restrict__ x,
                                     float* __restrict__ sq,
                                     float* __restrict__ out)
{
    if (blockIdx.x == 0 && threadIdx.x < 2) out[threadIdx.x] = 0.0f;

    const int lane = threadIdx.x & 31;
    const int wave = threadIdx.x >> 5;
    const int gw   = blockIdx.x * 8 + wave;      // row index == b*MM + m
    const float* row = x + (size_t)gw * FF;

    float s = 0.0f;
#pragma unroll
    for (int t = 0; t < FF / 32; ++t) {
        float v = row[lane + t * 32];
        s = fmaf(v, v, s);
    }
#pragma unroll
    for (int off = 16; off > 0; off >>= 1)
        s += __shfl_xor(s, off, 32);

    if (lane == 0) sq[gw] = s;
}

// ---------------------------------------------------------------------------
// Kernel 2: per part m, triangular tiled gram via V_WMMA_F32_16X16X4_F32,
// fused distance + mask + reduction epilogue.
// Block tile: 64 rows x 128 cols, 8 waves, each wave a 32x32 register tile.
// ---------------------------------------------------------------------------
__global__ __launch_bounds__(256, 2)
void MetricLoss_gram_kernel(const float* __restrict__ x,
                            const float* __restrict__ sq,
                            float* __restrict__ out)
{
    const int m     = blockIdx.z;        // part 0..31
    const int ibase = blockIdx.y * 64;   // 16 row blocks
    const int jbase = blockIdx.x * 128;  // 8  col blocks

    // All useful pairs have j > i: skip blocks entirely at/below diagonal.
    if (jbase + 127 <= ibase) return;

    const int tid   = threadIdx.x;
    const int lane  = tid & 31;
    const int w     = tid >> 5;
    const int wr    = w >> 2;            // 0..1
    const int wc    = w & 3;             // 0..3
    const int itile = ibase + wr * 32;
    const int jtile = jbase + wc * 32;
    if (jtile + 31 <= itile) return;     // wave-uniform: EXEC stays all-ones

    const int r  = lane & 15;            // row/col within 16-tile
    const int hs = lane >> 4;            // half select (K sub-offset)
    const int ko = hs * 2;

    // x[b][m][f] row pointers; rows are 1 KB contiguous.
    const float* a0 = x + ((size_t)(itile + r     ) * MM + m) * FF;
    const float* a1 = x + ((size_t)(itile + 16 + r) * MM + m) * FF;
    const float* b0 = x + ((size_t)(jtile + r     ) * MM + m) * FF;
    const float* b1 = x + ((size_t)(jtile + 16 + r) * MM + m) * FF;

    v8f acc00 = {}, acc01 = {}, acc10 = {}, acc11 = {};

    for (int k = 0; k < FF; k += 4) {
        v2f av0 = *(const v2f*)(a0 + k + ko);
        v2f av1 = *(const v2f*)(a1 + k + ko);
        v2f bv0 = *(const v2f*)(b0 + k + ko);
        v2f bv1 = *(const v2f*)(b1 + k + ko);
        acc00 = __builtin_amdgcn_wmma_f32_16x16x4_f32(false, av0, false, bv0,
                                                      (short)0, acc00, false, false);
        acc01 = __builtin_amdgcn_wmma_f32_16x16x4_f32(false, av0, false, bv1,
                                                      (short)0, acc01, false, false);
        acc10 = __builtin_amdgcn_wmma_f32_16x16x4_f32(false, av1, false, bv0,
                                                      (short)0, acc10, false, false);
        acc11 = __builtin_amdgcn_wmma_f32_16x16x4_f32(false, av1, false, bv1,
                                                      (short)0, acc11, false, false);
    }

    // Epilogue: d = sq_i + sq_j - 2*gram, masked accumulation.
    float homo = 0.0f, heter = 0.0f;
    auto process = [&](const v8f& acc, int it, int jt) {
        const int   j   = jt + r;                       // C/D: N = lane&15
        const float sqj = sq[(size_t)j * MM + m];
        const int   gj  = j >> 2;
#pragma unroll
        for (int rr = 0; rr < 8; ++rr) {
            const int   i   = it + rr + hs * 8;         // C/D: M = vgpr + 8*half
            const float sqi = sq[(size_t)i * MM + m];
            const float d   = sqi + sqj - 2.0f * acc[rr];
            const int   gi  = i >> 2;
            if (gi == gj) {
                if (i < j) homo += d;
            } else if (gi < gj) {
                heter += fmaxf(1.0f - d, 0.0f);
            }
        }
    };
    process(acc00, itile,      jtile);
    process(acc01, itile,      jtile + 16);
    process(acc10, itile + 16, jtile);
    process(acc11, itile + 16, jtile + 16);

#pragma unroll
    for (int off = 16; off > 0; off >>= 1) {
        homo  += __shfl_xor(homo,  off, 32);
        heter += __shfl_xor(heter, off, 32);
    }

    if (lane == 0) {
        constexpr float SC_HOMO  = 2.0f / (float)(BB * (KGRP - 1));
        constexpr float SC_HETER = 2.0f / (float)((size_t)BB * (BB - KGRP));
        atomicAdd(&out[0], homo  * SC_HOMO);
        atomicAdd(&out[1], heter * SC_HETER);
    }
}

// ---------------------------------------------------------------------------
extern "C" void kernel_launch(void* const* d_in, const int* in_sizes, int n_in,
                              void* d_out, int out_size, void* d_ws, size_t ws_size,
                              hipStream_t stream)
{
    (void)in_sizes; (void)n_in; (void)out_size; (void)ws_size;
    const float* x   = (const float*)d_in[0];
    float*       out = (float*)d_out;           // [loss_homo, loss_heter]
    float*       sq  = (float*)d_ws;            // BB*MM floats = 128 KB

    // 32768 rows, 8 waves (256 threads) per block -> 4096 blocks.
    MetricLoss_sq_kernel<<<(BB * MM) / 8, 256, 0, stream>>>(x, sq, out);

    dim3 grid(BB / 128, BB / 64, MM);           // (8, 16, 32)
    MetricLoss_gram_kernel<<<grid, 256, 0, stream>>>(x, sq, out);
}